// MXFP4ActivationQuantizer_88175678587515
// MI455X (gfx1250) — compile-verified
//
#include <hip/hip_runtime.h>
#include <math.h>
#include <stdint.h>

// ---------------------------------------------------------------------------
// MXFP4-style activation quantizer for (4, 4096, 4096) f32, GROUP_SIZE = 32.
// Scales are global per column-group: only 128 scales for the whole tensor.
//   pass 1: max|x| per column-group (uint-bit atomicMax trick); input streamed
//           through LDS with gfx1250 async global->LDS loads (4-deep ring)
//   pass 2: exps = floor(log2(max)) (exact via frexpf), scale tables
//   pass 3: q = sign(x) * 2^(e-3) * rint(min(|x|*2^-e, 1) * 8)
// Memory-bound: ~768 MB total traffic -> ~33us at 23.3 TB/s.
// ---------------------------------------------------------------------------

#ifndef __has_builtin
#define __has_builtin(x) 0
#endif

// gfx1250 async global->LDS path (probe-confirmed present on this toolchain:
// arg0 is v4i addrspace(1)*, per the round-1 diagnostic).
#if defined(__HIP_DEVICE_COMPILE__)
#if __has_builtin(__builtin_amdgcn_global_load_async_to_lds_b128) && \
    __has_builtin(__builtin_amdgcn_s_wait_asynccnt)
#define MXQ_USE_ASYNC 1
#endif
#endif

#define MXQ_D    4096            // last-dim length (per reference)
#define MXQ_C4   (MXQ_D / 4)     // 1024 float4 per row
#define MXQ_G    128             // groups per row (4096/32)
#define MXQ_BLK  256             // threads per block
#define MXQ_RPB  64              // rows per block (grid.y)

// d_ws layout (32-bit slots):
//   [  0..127]  uint  : max|x| bit pattern per group
//   [128..255]  float : 2^(e-3)   (scale/8)
//   [256..383]  float : 2^-e      (inverse scale)

__global__ void mxq_init(unsigned int* __restrict__ wsu) {
    int g = threadIdx.x;
    if (g < MXQ_G) wsu[g] = 0u;
}

#if MXQ_USE_ASYNC
typedef int mxq_v4i __attribute__((ext_vector_type(4)));
typedef __attribute__((address_space(1))) mxq_v4i* mxq_gp;   // global int4*
typedef __attribute__((address_space(3))) mxq_v4i* mxq_lp;   // LDS int4*

__device__ __forceinline__ void mxq_async_b128(const void* gptr, unsigned lds_off) {
    // (global src, lds dst, imm offset, cpol); uintptr_t round-trip legally
    // changes address space and drops const.
    __builtin_amdgcn_global_load_async_to_lds_b128(
        (mxq_gp)(uintptr_t)gptr, (mxq_lp)(uintptr_t)lds_off, 0, 0);
}
#endif

__device__ __forceinline__ float mxq_fmax4(float m, float4 a) {
    return fmaxf(m, fmaxf(fmaxf(fabsf(a.x), fabsf(a.y)),
                          fmaxf(fabsf(a.z), fabsf(a.w))));
}

__global__ __launch_bounds__(MXQ_BLK) void mxq_reduce(const float4* __restrict__ act4,
                                                      unsigned int* __restrict__ wsu) {
    const int c4 = blockIdx.x * MXQ_BLK + threadIdx.x;   // fixed float4 column
    const int r0 = blockIdx.y * MXQ_RPB;
    const float4* p = act4 + (size_t)r0 * MXQ_C4 + c4;
    float m = 0.0f;

#if MXQ_USE_ASYNC
    // 4-deep LDS ring buffer (16 KB/block). Async loads complete in order,
    // so s_wait_asynccnt(k) means "all but the newest k are done".
    __shared__ float4 stage[4][MXQ_BLK];
    unsigned loff[4];
    #pragma unroll
    for (int i = 0; i < 4; ++i)
        loff[i] = (unsigned)(uintptr_t)&stage[i][threadIdx.x];

    #pragma unroll
    for (int i = 0; i < 4; ++i)                          // prologue: rows 0..3
        mxq_async_b128(p + (size_t)i * MXQ_C4, loff[i]);

    for (int r = 0; r < MXQ_RPB - 4; ++r) {
        __builtin_amdgcn_s_wait_asynccnt(3);             // row r has landed
        float4 a = stage[r & 3][threadIdx.x];
        m = mxq_fmax4(m, a);
        mxq_async_b128(p + (size_t)(r + 4) * MXQ_C4, loff[r & 3]);
    }
    {                                                    // drain rows RPB-4..RPB-1
        const int r = MXQ_RPB - 4;
        __builtin_amdgcn_s_wait_asynccnt(3);
        m = mxq_fmax4(m, stage[(r + 0) & 3][threadIdx.x]);
        __builtin_amdgcn_s_wait_asynccnt(2);
        m = mxq_fmax4(m, stage[(r + 1) & 3][threadIdx.x]);
        __builtin_amdgcn_s_wait_asynccnt(1);
        m = mxq_fmax4(m, stage[(r + 2) & 3][threadIdx.x]);
        __builtin_amdgcn_s_wait_asynccnt(0);
        m = mxq_fmax4(m, stage[(r + 3) & 3][threadIdx.x]);
    }
#else
    const bool pf_ok = (blockIdx.y + 1 < gridDim.y);
    #pragma unroll 4
    for (int r = 0; r < MXQ_RPB; ++r) {
        float4 a = p[(size_t)r * MXQ_C4];
        if (pf_ok) __builtin_prefetch(p + (size_t)(r + 8) * MXQ_C4, 0, 0);
        m = mxq_fmax4(m, a);
    }
#endif
    // |x| >= 0, so float bit patterns order like unsigned ints.
    atomicMax(&wsu[c4 >> 3], __float_as_uint(m));
}

__global__ void mxq_exps(const unsigned int* __restrict__ wsu,
                         float* __restrict__ wsf,
                         float* __restrict__ exps_out) {
    int g = threadIdx.x;
    if (g >= MXQ_G) return;
    float m = __uint_as_float(wsu[g]);
    float e = 0.0f;
    if (m > 0.0f) {
        int ex;
        (void)frexpf(m, &ex);       // m = f * 2^ex, f in [0.5, 1)
        e = (float)(ex - 1);        // exact floor(log2(m))
    }
    exps_out[g] = e;
    int ei = (int)e;
    wsf[128 + g] = ldexpf(1.0f, ei - 3);  // scale / 8 (exact power of two)
    wsf[256 + g] = ldexpf(1.0f, -ei);     // 1 / scale (exact power of two)
}

__device__ __forceinline__ float mxq_q1(float x, float s8, float inv) {
    float t = fminf(fabsf(x) * inv, 1.0f);   // clip(|x|/scale, 0, 1)
    float k = rintf(t * 8.0f);               // round-half-even, matches jnp.round
    return copysignf(s8 * k, x);             // sign * scale * (k/8), exact pow2 scaling
}

__global__ __launch_bounds__(MXQ_BLK) void mxq_quant(const float4* __restrict__ act4,
                                                     float4* __restrict__ out4,
                                                     const float* __restrict__ wsf) {
    const int c4 = blockIdx.x * MXQ_BLK + threadIdx.x;
    const int g  = c4 >> 3;                  // 8 float4 columns per group of 32
    const float s8  = wsf[128 + g];
    const float inv = wsf[256 + g];
    const int r0 = blockIdx.y * MXQ_RPB;
    const size_t base = (size_t)r0 * MXQ_C4 + c4;
    const bool pf_ok = (blockIdx.y + 1 < gridDim.y);

    #pragma unroll 4
    for (int r = 0; r < MXQ_RPB; ++r) {
        float4 a = act4[base + (size_t)r * MXQ_C4];
        if (pf_ok) __builtin_prefetch(&act4[base + (size_t)(r + 8) * MXQ_C4], 0, 0);
        float4 q;
        q.x = mxq_q1(a.x, s8, inv);
        q.y = mxq_q1(a.y, s8, inv);
        q.z = mxq_q1(a.z, s8, inv);
        q.w = mxq_q1(a.w, s8, inv);
        out4[base + (size_t)r * MXQ_C4] = q;
    }
}

extern "C" void kernel_launch(void* const* d_in, const int* in_sizes, int n_in,
                              void* d_out, int out_size, void* d_ws, size_t ws_size,
                              hipStream_t stream) {
    const float* act = (const float*)d_in[0];
    const size_t n = (size_t)in_sizes[0];        // 4*4096*4096 = 67108864
    const int rows = (int)(n / MXQ_D);           // 16384

    float*        q_out    = (float*)d_out;
    float*        exps_out = (float*)d_out + n;  // tuple tail: 128 exps
    unsigned int* wsu      = (unsigned int*)d_ws;
    float*        wsf      = (float*)d_ws;

    dim3 blk(MXQ_BLK);
    dim3 grd(MXQ_C4 / MXQ_BLK, rows / MXQ_RPB);  // (4, 256) = 1024 blocks

    mxq_init  <<<1, 128, 0, stream>>>(wsu);
    mxq_reduce<<<grd, blk, 0, stream>>>((const float4*)act, wsu);
    mxq_exps  <<<1, 128, 0, stream>>>(wsu, wsf, exps_out);
    mxq_quant <<<grd, blk, 0, stream>>>((const float4*)act, (float4*)q_out, wsf);
}